// StyleTransferLoss_39917426049458
// MI455X (gfx1250) — compile-verified
//
#include <hip/hip_runtime.h>
#include <hip/hip_bf16.h>

#define NBINS 256
#define IMG_H 512
#define IMG_W 512
#define HW (IMG_H * IMG_W)          // 262144 pixels per plane
#define NBATCH 32
#define NCH 3
#define CHUNK 4096                  // pixels per block in streaming kernels
#define NCHUNK (HW / CHUNK)         // 64 blocks per image

typedef float v2f __attribute__((ext_vector_type(2)));
typedef float v8f __attribute__((ext_vector_type(8)));

__device__ __forceinline__ float de_norm255(float x) {
    float f = (x + 1.0f) * 0.5f;
    f = fminf(fmaxf(f, 0.0f), 1.0f);
    return f * 255.0f;
}

__device__ __forceinline__ int bin_of(float v) {
    // idx = clip(floor(v * 256/255), 0, 255)
    int idx = (int)floorf(v * (256.0f / 255.0f));
    idx = idx < 0 ? 0 : idx;
    return idx > (NBINS - 1) ? (NBINS - 1) : idx;
}

// ---------------------------------------------------------------------------
// Kernel 1: mask-weighted histograms of de-normed final & target.
// g_hist layout: [b][2][3][256] f32   (tsel 0 = final, 1 = target)
// ---------------------------------------------------------------------------
__global__ void hist_kernel(const float* __restrict__ final_,
                            const float* __restrict__ target_,
                            const int*   __restrict__ mask_,
                            float*       __restrict__ g_hist) {
    __shared__ float lh[6][NBINS];               // [tsel*3+c][bin]
    const int t  = threadIdx.x;                  // 0..255
    const int b  = blockIdx.y;
    const int p0 = blockIdx.x * CHUNK;

    #pragma unroll
    for (int k = 0; k < 6; ++k) lh[k][t] = 0.0f;
    __syncthreads();

    const int4* m4 = (const int4*)(mask_ + (size_t)b * HW + p0);
    const float4* f4[NCH];
    const float4* g4[NCH];
    #pragma unroll
    for (int c = 0; c < NCH; ++c) {
        f4[c] = (const float4*)(final_  + ((size_t)(b * NCH + c)) * HW + p0);
        g4[c] = (const float4*)(target_ + ((size_t)(b * NCH + c)) * HW + p0);
    }

    for (int it = 0; it < CHUNK / (256 * 4); ++it) {
        const int i4 = it * 256 + t;
        const int4 m = m4[i4];
        float fc[NCH][4], tc[NCH][4];
        #pragma unroll
        for (int c = 0; c < NCH; ++c) {
            float4 fv = f4[c][i4];
            float4 tv = g4[c][i4];
            fc[c][0] = fv.x; fc[c][1] = fv.y; fc[c][2] = fv.z; fc[c][3] = fv.w;
            tc[c][0] = tv.x; tc[c][1] = tv.y; tc[c][2] = tv.z; tc[c][3] = tv.w;
        }
        const int me[4] = {m.x, m.y, m.z, m.w};
        #pragma unroll
        for (int e = 0; e < 4; ++e) {
            if (me[e] != 0) {
                const float w = (float)me[e];
                #pragma unroll
                for (int c = 0; c < NCH; ++c) {
                    const float fm = de_norm255(fc[c][e]) * w;   // img_m
                    const float tm = de_norm255(tc[c][e]) * w;   // tar_m
                    atomicAdd(&lh[c][bin_of(fm)], w);
                    atomicAdd(&lh[3 + c][bin_of(tm)], w);
                }
            }
        }
    }
    __syncthreads();

    float* gh = g_hist + (size_t)b * 6 * NBINS;
    #pragma unroll
    for (int k = 0; k < 6; ++k) {
        const float v = lh[k][t];
        if (v != 0.0f) atomicAdd(&gh[k * NBINS + t], v);
    }
}

// ---------------------------------------------------------------------------
// Kernel 2: per (b,c) CDFs + transfer table.
// 256-bin inclusive cumsum done on the matrix pipe with V_WMMA_F32_16X16X4_F32:
//   view hist as H[16][16]; row-wise prefix P = H x U (U upper-tri ones),
//   accumulated over 4 K=4 chunks; inter-row offsets added afterwards.
// Exact in f32 (integer counts < 2^24).
// ---------------------------------------------------------------------------
__global__ void table_kernel(const float* __restrict__ g_hist,
                             float*       __restrict__ g_table) {
    __shared__ float h0[NBINS], h1[NBINS];       // hist img / tar
    __shared__ float q0[NBINS], q1[NBINS];       // row-prefix results
    __shared__ float c0[NBINS], c1[NBINS];       // normalized CDFs
    __shared__ float red[NBINS];
    __shared__ float cntS;

    const int t  = threadIdx.x;                  // 0..255
    const int bc = blockIdx.x;                   // b*3 + c
    const int b  = bc / NCH;
    const int c  = bc % NCH;

    const float* gh = g_hist + (size_t)b * 6 * NBINS;
    h0[t] = gh[c * NBINS + t];
    h1[t] = gh[(3 + c) * NBINS + t];
    red[t] = h0[t];                              // cnt = sum(mask) = sum(hist)
    __syncthreads();
    for (int s = 128; s > 0; s >>= 1) {
        if (t < s) red[t] += red[t + s];
        __syncthreads();
    }
    if (t == 0) cntS = red[0];
    __syncthreads();
    const float cnt = cntS;

    const int wave = t >> 5;
    const int lane = t & 31;
    if (wave < 2) {                              // wave-uniform: EXEC all ones
        const float* h = (wave == 0) ? h0 : h1;
        float*       q = (wave == 0) ? q0 : q1;
        const int r  = lane & 15;                // A row, B/D col index
        const int hi = lane >> 4;                // lane half
        v8f acc = {};
        #pragma unroll
        for (int kc = 0; kc < 4; ++kc) {
            // A 16x4 (ISA 7.12.2): lanes 0-15 K=4kc+{0,1}, lanes 16-31 K=4kc+{2,3}
            v2f a;
            const int ka = 4 * kc + 2 * hi;
            a.x = h[16 * r + ka];
            a.y = h[16 * r + ka + 1];
            // B 4x16 upper-triangular ones; VGPR v holds row K = v + 2*hi, col N = lane&15
            v2f bf;
            bf.x = ((4 * kc + 2 * hi + 0) <= r) ? 1.0f : 0.0f;
            bf.y = ((4 * kc + 2 * hi + 1) <= r) ? 1.0f : 0.0f;
            acc = __builtin_amdgcn_wmma_f32_16x16x4_f32(
                false, a, false, bf, (short)0, acc, false, false);
        }
        // D 16x16: VGPR v -> M = v + 8*hi, N = lane&15
        #pragma unroll
        for (int v = 0; v < 8; ++v) q[16 * (v + 8 * hi) + r] = acc[v];
    }
    __syncthreads();

    {   // add exclusive prefix of row sums, normalize by cnt
        const int row = t >> 4;
        float off0 = 0.0f, off1 = 0.0f;
        for (int r = 0; r < row; ++r) {
            off0 += q0[16 * r + 15];
            off1 += q1[16 * r + 15];
        }
        const float inv = (cnt > 0.0f) ? (1.0f / cnt) : 0.0f;
        c0[t] = (q0[t] + off0) * inv;            // dst_cdf (final)
        c1[t] = (q1[t] + off1) * inv;            // ref_cdf (target)
    }
    __syncthreads();

    {   // cal_trans: first j with dst in [ref[j], ref[j+1]] -> j+1 else identity
        const float rv = c0[t];
        int tj = t;
        for (int j = 0; j < NBINS - 1; ++j) {
            if (rv >= c1[j] && rv <= c1[j + 1]) { tj = j + 1; break; }
        }
        if (t == 0) tj = 0;
        if (t == NBINS - 1) tj = NBINS - 1;
        g_table[(size_t)bc * NBINS + t] = (float)tj;
    }
}

// ---------------------------------------------------------------------------
// Kernel 3: sum |img_m - table[int(img_m)]| over masked pixels (f64 atomic).
// ---------------------------------------------------------------------------
__global__ void loss_kernel(const float* __restrict__ final_,
                            const int*   __restrict__ mask_,
                            const float* __restrict__ g_table,
                            double*      __restrict__ acc) {
    __shared__ float red[256];
    const int t  = threadIdx.x;
    const int b  = blockIdx.y;
    const int p0 = blockIdx.x * CHUNK;

    const int4* m4 = (const int4*)(mask_ + (size_t)b * HW + p0);
    const float4* f4[NCH];
    #pragma unroll
    for (int c = 0; c < NCH; ++c)
        f4[c] = (const float4*)(final_ + ((size_t)(b * NCH + c)) * HW + p0);
    const float* tab = g_table + (size_t)b * NCH * NBINS;

    float local = 0.0f;
    for (int it = 0; it < CHUNK / (256 * 4); ++it) {
        const int i4 = it * 256 + t;
        const int4 m = m4[i4];
        float fc[NCH][4];
        #pragma unroll
        for (int c = 0; c < NCH; ++c) {
            float4 fv = f4[c][i4];
            fc[c][0] = fv.x; fc[c][1] = fv.y; fc[c][2] = fv.z; fc[c][3] = fv.w;
        }
        const int me[4] = {m.x, m.y, m.z, m.w};
        #pragma unroll
        for (int e = 0; e < 4; ++e) {
            if (me[e] != 0) {
                const float w = (float)me[e];
                #pragma unroll
                for (int c = 0; c < NCH; ++c) {
                    const float fm = de_norm255(fc[c][e]) * w;   // img_m
                    int pix = (int)fm;                            // trunc toward 0, fm >= 0
                    pix = pix > (NBINS - 1) ? (NBINS - 1) : pix;
                    local += fabsf(fm - tab[c * NBINS + pix]);
                }
            }
        }
    }

    red[t] = local;
    __syncthreads();
    for (int s = 128; s > 0; s >>= 1) {
        if (t < s) red[t] += red[t + s];
        __syncthreads();
    }
    if (t == 0) atomicAdd(acc, (double)red[0]);
}

__global__ void finalize_kernel(const double* __restrict__ acc,
                                float* __restrict__ out) {
    if (threadIdx.x == 0 && blockIdx.x == 0)
        out[0] = (float)(acc[0] * (1.0 / ((double)NCH * (double)HW)));
}

// ---------------------------------------------------------------------------
extern "C" void kernel_launch(void* const* d_in, const int* in_sizes, int n_in,
                              void* d_out, int out_size, void* d_ws, size_t ws_size,
                              hipStream_t stream) {
    (void)in_sizes; (void)n_in; (void)out_size; (void)ws_size;
    const float* d_final  = (const float*)d_in[0];
    const float* d_target = (const float*)d_in[1];
    const int*   d_mask   = (const int*)d_in[2];
    float* out = (float*)d_out;

    char* ws = (char*)d_ws;
    double* acc    = (double*)ws;                                  // 8 B (pad 256)
    float*  g_hist = (float*)(ws + 256);                           // 32*6*256 f32
    float*  g_table = (float*)(ws + 256 + (size_t)NBATCH * 6 * NBINS * sizeof(float));

    // zero accumulator + histograms (graph-capture safe)
    hipMemsetAsync(ws, 0, 256 + (size_t)NBATCH * 6 * NBINS * sizeof(float), stream);

    dim3 grid(NCHUNK, NBATCH);
    hist_kernel<<<grid, 256, 0, stream>>>(d_final, d_target, d_mask, g_hist);
    table_kernel<<<NBATCH * NCH, 256, 0, stream>>>(g_hist, g_table);
    loss_kernel<<<grid, 256, 0, stream>>>(d_final, d_mask, g_table, acc);
    finalize_kernel<<<1, 32, 0, stream>>>(acc, out);
}